// GINE_MultiLayer_58007828300389
// MI455X (gfx1250) — compile-verified
//
#include <hip/hip_runtime.h>

// ---------------------------------------------------------------------------
// GINE 3-layer GNN for MI455X (gfx1250, wave32).
//
// Roofline: edge gather/scatter ~2.7 GB fp32 traffic dominates (~120us @
// 23.3 TB/s); GEMMs are ~14.7 GFLOP fp32 -> WMMA pipe via
// V_WMMA_F32_16X16X4_F32 (exact fp32 semantics, matches the reference).
// edge_attr (307 MB/pass, single use) is loaded non-temporally so the L2
// keeps the randomly-gathered x rows (25.6 MB, high reuse) instead.
// GEMM A-tiles are staged global->LDS with the gfx1250 async-LDS path
// (ASYNCcnt) when the builtin is available.
// ---------------------------------------------------------------------------

typedef float v2f __attribute__((ext_vector_type(2)));
typedef float v4f __attribute__((ext_vector_type(4)));
typedef float v8f __attribute__((ext_vector_type(8)));
typedef int   v4i __attribute__((ext_vector_type(4)));

#define N_NODES 50000
#define N_EDGES 600000
#define DIM     128

#if defined(__AMDGCN__) && __has_builtin(__builtin_amdgcn_global_load_async_to_lds_b128)
#define USE_ASYNC_LDS 1
typedef __attribute__((address_space(1))) v4i GV4;   // global (AS1) int4
typedef __attribute__((address_space(3))) v4i LV4;   // LDS (AS3) int4
#else
#define USE_ASYNC_LDS 0
#endif

// h[node][0:128] = x[node][0:128]   (x rows may be strided: 128 or 384)
__global__ __launch_bounds__(256)
void copy_rows_kernel(const float* __restrict__ x, int ldx,
                      float* __restrict__ h)
{
    int idx  = blockIdx.x * blockDim.x + threadIdx.x;   // one float4 each
    int node = idx >> 5;                                // 32 float4 per row
    int j    = idx & 31;
    if (node >= N_NODES) return;
    const v4f* xp = (const v4f*)(x + (size_t)node * ldx);
    v4f*       hp = (v4f*)(h + (size_t)node * DIM);
    hp[j] = xp[j];
}

// One wave per edge: msg = relu(x[src] + edge_attr); atomic-add into h[dst].
__global__ __launch_bounds__(256)
void edge_scatter_kernel(const float* __restrict__ x, int ldx,
                         const float* __restrict__ ea,
                         const int* __restrict__ src,
                         const int* __restrict__ dst,
                         float* __restrict__ h)
{
    int e = blockIdx.x * (blockDim.x >> 5) + (threadIdx.x >> 5);
    if (e >= N_EDGES) return;
    int lane = threadIdx.x & 31;

    int s = src[e];
    int d = dst[e];

    v4f xv = ((const v4f*)(x + (size_t)s * ldx))[lane];              // cached: reused
    v4f ev = __builtin_nontemporal_load(                              // NT: single use
                 (const v4f*)(ea + (size_t)e * DIM) + lane);

    v4f m;
    m.x = fmaxf(xv.x + ev.x, 0.0f);
    m.y = fmaxf(xv.y + ev.y, 0.0f);
    m.z = fmaxf(xv.z + ev.z, 0.0f);
    m.w = fmaxf(xv.w + ev.w, 0.0f);

    float* hp = h + (size_t)d * DIM + lane * 4;
    unsafeAtomicAdd(hp + 0, m.x);   // hardware global_atomic_add_f32 (no-return)
    unsafeAtomicAdd(hp + 1, m.y);
    unsafeAtomicAdd(hp + 2, m.z);
    unsafeAtomicAdd(hp + 3, m.w);
}

// Y[rt*16 : rt*16+16, 0:128] = relu( X_tile[16,K] @ W[K,128] + bias )
// X rows are contiguous (ldx == K).  One block per 16-row tile; the tile is
// staged in LDS (async global->LDS when available); 8 waves each own one
// 16-column output tile.  fp32 WMMA fragment layout (wave32):
//   A 16x4: lane L (L<16) row=L holds K=4k+{0,1}; lane L+16 holds K=4k+{2,3}
//   B 4x16: lane L (L<16) col=L holds K rows {0,1}; lane L+16 holds {2,3}
//   D 16x16: VGPR i -> row i (lanes 0-15) / row i+8 (lanes 16-31), col=lane%16
template<int K>
__global__ __launch_bounds__(256)
void gemm_relu_wmma_kernel(const float* __restrict__ X,
                           const float* __restrict__ W,     // [K,128] row-major
                           const float* __restrict__ bias,  // [128]
                           float* __restrict__ Y, int ldy)
{
    __shared__ float sx[16 * K];

    int rt = blockIdx.x;                              // row tile (16 rows)
    const float* xt = X + (size_t)rt * 16 * K;
    constexpr int total4 = (16 * K) >> 2;             // float4 chunks in tile

#if USE_ASYNC_LDS
    for (int i = threadIdx.x; i < total4; i += 256) {
        __builtin_amdgcn_global_load_async_to_lds_b128(
            (GV4*)((const v4f*)xt + i),               // global src (AS1)
            (LV4*)((v4f*)sx + i),                     // LDS dst   (AS3)
            /*offset=*/0, /*cpol=*/0);
    }
    asm volatile("s_wait_asynccnt 0x0" ::: "memory");
    __syncthreads();
#else
    for (int i = threadIdx.x; i < total4; i += 256)
        ((v4f*)sx)[i] = ((const v4f*)xt)[i];
    __syncthreads();
#endif

    int lane  = threadIdx.x & 31;
    int wv    = threadIdx.x >> 5;                     // col tile 0..7
    int l16   = lane & 15;
    int khalf = (lane >> 4) << 1;                     // 0 or 2
    int col   = wv * 16 + l16;

    v8f acc = {0.f, 0.f, 0.f, 0.f, 0.f, 0.f, 0.f, 0.f};
    const float* srow = sx + l16 * K;

#pragma unroll 8
    for (int kb = 0; kb < K; kb += 4) {
        int kk = kb + khalf;
        float2 av = *(const float2*)(srow + kk);      // ds_load_b64
        v2f a = {av.x, av.y};
        v2f b = {W[(size_t)kk * DIM + col],
                 W[(size_t)(kk + 1) * DIM + col]};
        acc = __builtin_amdgcn_wmma_f32_16x16x4_f32(
                  /*neg_a=*/false, a, /*neg_b=*/false, b,
                  /*c_mod=*/(short)0, acc,
                  /*reuse_a=*/false, /*reuse_b=*/false);
    }

    float bv = bias[col];
    int rbase = rt * 16 + ((lane >> 4) << 3);         // +8 for upper half-wave
    for (int i = 0; i < 8; ++i) {
        float r = fmaxf(acc[i] + bv, 0.0f);           // fused bias + ReLU
        Y[(size_t)(rbase + i) * ldy + col] = r;
    }
}

extern "C" void kernel_launch(void* const* d_in, const int* in_sizes, int n_in,
                              void* d_out, int out_size, void* d_ws, size_t ws_size,
                              hipStream_t stream)
{
    const float* x0 = (const float*)d_in[0];
    const int*   ei = (const int*)d_in[1];          // [2, E] int32
    const float* ea = (const float*)d_in[2];
    const float* W1[3] = {(const float*)d_in[3],  (const float*)d_in[7],  (const float*)d_in[11]};
    const float* b1[3] = {(const float*)d_in[4],  (const float*)d_in[8],  (const float*)d_in[12]};
    const float* W2[3] = {(const float*)d_in[5],  (const float*)d_in[9],  (const float*)d_in[13]};
    const float* b2[3] = {(const float*)d_in[6],  (const float*)d_in[10], (const float*)d_in[14]};
    const float* Wc = (const float*)d_in[15];
    const float* bc = (const float*)d_in[16];
    const int* src = ei;
    const int* dst = ei + N_EDGES;

    float* ws = (float*)d_ws;
    size_t nf = (size_t)N_NODES * DIM;
    float* h   = ws;                 // [N,128] accumulator: x + scatter(msg)
    float* tmp = ws + nf;            // [N,128] after Linear1+ReLU
    float* xs  = ws + 2 * nf;        // [N,384] JK concat buffer

    dim3 blk(256);
    int copyGrid = (N_NODES * 32) / 256;   // 6250  (exact)
    int edgeGrid = N_EDGES / 8;            // 75000 (exact, 8 waves/block)
    int gemmGrid = N_NODES / 16;           // 3125  (exact, EXEC all-ones)

    const float* xcur = x0;
    int ldx = DIM;
    for (int l = 0; l < 3; ++l) {
        copy_rows_kernel<<<copyGrid, blk, 0, stream>>>(xcur, ldx, h);
        edge_scatter_kernel<<<edgeGrid, blk, 0, stream>>>(xcur, ldx, ea, src, dst, h);
        gemm_relu_wmma_kernel<DIM><<<gemmGrid, blk, 0, stream>>>(h,   W1[l], b1[l], tmp,          DIM);
        gemm_relu_wmma_kernel<DIM><<<gemmGrid, blk, 0, stream>>>(tmp, W2[l], b2[l], xs + l * DIM, 3 * DIM);
        xcur = xs + l * DIM;               // next layer reads its slice of xs
        ldx  = 3 * DIM;
    }
    // JK 'cat' + lin_cat: [N,384] @ [384,128] + bc, ReLU
    gemm_relu_wmma_kernel<3 * DIM><<<gemmGrid, blk, 0, stream>>>(xs, Wc, bc, (float*)d_out, DIM);
}